// Autoencoder_84911503442553
// MI455X (gfx1250) — compile-verified
//
#include <hip/hip_runtime.h>
#include <math.h>

typedef __attribute__((ext_vector_type(2))) float v2f;
typedef __attribute__((ext_vector_type(8))) float v8f;
typedef unsigned int v4u __attribute__((ext_vector_type(4)));
typedef int v8i __attribute__((ext_vector_type(8)));
typedef int v4i __attribute__((ext_vector_type(4)));

#define BB   4
#define LL   2048
#define HH   8
#define DD   64
#define SK   40
#define NTOP 40
#define PADU 48   // NTOP padded to 3 row-tiles of 16
#define VTS  18   // padded LDS stride for transposed V (even -> b64 aligned, bank-spread)

// ---- lane permutes on the VALU (no LDS traffic) ----
__device__ __forceinline__ float xhalf16(float x) {  // exchange with lane^16
  int xi = __float_as_int(x);
  int r = __builtin_amdgcn_permlanex16(xi, xi, 0x76543210, 0xFEDCBA98, false, false);
  return __int_as_float(r);
}
__device__ __forceinline__ float pl16(float x, int lo, int hi) {
  int xi = __float_as_int(x);
  int r = __builtin_amdgcn_permlane16(xi, xi, lo, hi, false, false);
  return __int_as_float(r);
}
__device__ __forceinline__ float wave_sum32(float x) {
  x += xhalf16(x);                                       // xor 16
  x += pl16(x, (int)0xFEDCBA98, (int)0x76543210);        // xor 8
  x += pl16(x, (int)0x32107654, (int)0xBA98FEDC);        // xor 4
  x += pl16(x, (int)0x54761032, (int)0xDCFE98BA);        // xor 2
  x += pl16(x, (int)0x67452301, (int)0xEFCDAB89);        // xor 1
  return x;
}

// ---- Tensor Data Mover: 2D tile (16 rows x 64 f32, row stride 512 elems) -> LDS ----
// D# per CDNA5 ISA ch.8: group0 = {count|flags, lds_addr, global_addr, type=2},
// group1 = {data_size=4B, tensor_dim0=64, tensor_dim1=2048, tile=64x16, stride0=512}.
__device__ __forceinline__ void tdm_load_k_tile(const float* gptr, unsigned lds_off) {
  unsigned long long ga = (unsigned long long)(size_t)gptr;
  v4u g0;
  g0[0] = 1u;                                            // count=1 (valid user D#)
  g0[1] = lds_off;                                       // lds_addr (bytes)
  g0[2] = (unsigned)(ga & 0xFFFFFFFFu);                  // global_addr[31:0]
  g0[3] = (unsigned)((ga >> 32) & 0x01FFFFFFu) | (2u << 30);  // addr[56:32] | type=2
  v8i g1;
  g1[0] = (int)(2u << 16);     // workgroup_mask=0 | data_size=2 (4 bytes)
  g1[1] = (int)(64u << 16);    // atomic_barrier_addr=0 | tensor_dim0 lo16 = 64
  g1[2] = (int)(2048u << 16);  // tensor_dim0 hi=0 | tensor_dim1 lo16 = 2048
  g1[3] = (int)(64u << 16);    // tensor_dim1 hi=0 | tile_dim0 = 64
  g1[4] = 16;                  // tile_dim1 = 16 | tile_dim2 = 0
  g1[5] = 512;                 // tensor_dim0_stride = 512 elements (H*D)
  g1[6] = 0;                   // stride0 hi | tensor_dim1_stride lo (unused, 2D)
  g1[7] = 0;
  v4i g2 = {0, 0, 0, 0};
  v4i g3 = {0, 0, 0, 0};
#if __clang_major__ >= 23
  v8i gz = {0, 0, 0, 0, 0, 0, 0, 0};
  __builtin_amdgcn_tensor_load_to_lds(g0, g1, g2, g3, gz, 0);
#else
  __builtin_amdgcn_tensor_load_to_lds(g0, g1, g2, g3, 0);
#endif
}

// ---------------- Phase 1: M = max_s(q.k_s) - mean_all ----------------
__global__ __launch_bounds__(256) void ps_measure_kernel(
    const float* __restrict__ q, const float* __restrict__ k,
    const int* __restrict__ idx, float* __restrict__ Mv) {
  int wid  = blockIdx.x * 8 + (threadIdx.x >> 5);
  int lane = threadIdx.x & 31;
  int l  = wid & (LL - 1);
  int bh = wid >> 11;
  int h  = bh & (HH - 1);
  int b  = bh >> 3;

  const float* qrow = q + ((size_t)(b * LL + l) * HH + h) * DD;
  float2 qv = *(const float2*)(qrow + lane * 2);

  float mmax = -INFINITY, msum = 0.f;
  #pragma unroll 4
  for (int s = 0; s < SK; ++s) {
    int ki = idx[l * SK + s];
    const float* krow = k + ((size_t)(b * LL + ki) * HH + h) * DD;
    float2 kv = *(const float2*)(krow + lane * 2);
    float p = qv.x * kv.x + qv.y * kv.y;
    p = wave_sum32(p);  // replicated full dot
    mmax = fmaxf(mmax, p);
    msum += p;
  }
  if (lane == 0) Mv[wid] = mmax - msum * (1.0f / (float)LL);
}

// ---------------- Phase 2: top-40 per (b,h) (top_k order), gather Q_reduce ----------------
__global__ __launch_bounds__(256) void ps_topk_gather_kernel(
    const float* __restrict__ Mv, const float* __restrict__ q,
    int* __restrict__ Mtop, float* __restrict__ Qred) {
  __shared__ float vals[LL];
  __shared__ float rv[256];
  __shared__ int   ri[256];
  __shared__ int   topIdx[NTOP];
  int tid = threadIdx.x;
  int bh  = blockIdx.x;
  int h   = bh & (HH - 1);
  int b   = bh >> 3;

  for (int i = tid; i < LL; i += 256) vals[i] = Mv[(size_t)bh * LL + i];
  __syncthreads();

  for (int t = 0; t < NTOP; ++t) {
    float bv = -INFINITY; int bi = LL;
    for (int i = tid; i < LL; i += 256) {
      float v = vals[i];
      if (v > bv || (v == bv && i < bi)) { bv = v; bi = i; }
    }
    rv[tid] = bv; ri[tid] = bi;
    __syncthreads();
    for (int s = 128; s > 0; s >>= 1) {
      if (tid < s) {
        float ov = rv[tid + s]; int oi = ri[tid + s];
        if (ov > rv[tid] || (ov == rv[tid] && oi < ri[tid])) { rv[tid] = ov; ri[tid] = oi; }
      }
      __syncthreads();
    }
    if (tid == 0) { topIdx[t] = ri[0]; vals[ri[0]] = -INFINITY; }
    __syncthreads();
  }

  if (tid < NTOP) Mtop[bh * NTOP + tid] = topIdx[tid];

  // Gather selected Q rows (pre-scaled by 1/sqrt(D): exact pow2), zero-pad [40,48)
  for (int i = tid; i < PADU * DD; i += 256) {
    int r = i >> 6, d = i & 63;
    float val = 0.f;
    if (r < NTOP) {
      int lq = topIdx[r];
      val = q[((size_t)(b * LL + lq) * HH + h) * DD + d] * 0.125f;
    }
    Qred[((size_t)bh * PADU + r) * DD + d] = val;
  }
}

// ---------------- Phase 3: flash attention, transposed formulation + TDM K prefetch ----------------
// S^T = K·Q^T  (A = K tile, B = Q regs), per-lane softmax (lane = query),
// O^T += V^T·P^T. K tiles double-buffered via tensor_load_to_lds (TENSORcnt),
// V staged manually (needs transpose).
__global__ __launch_bounds__(96) void ps_attn_wmma_kernel(
    const float* __restrict__ kg, const float* __restrict__ vg,
    const float* __restrict__ Qred, float* __restrict__ out) {
  __shared__ float kT[2 * 16 * DD];     // double-buffered [key][d] 16 x 64
  __shared__ float vt[DD * VTS];        // [d][key]   64 x 18 (padded)
  __shared__ float pbuf[3 * 16 * 16];   // per-wave P^T as [q][key]

  int tid  = threadIdx.x;
  int w    = tid >> 5;
  int lane = tid & 31;
  int half = lane >> 4;
  int ln   = lane & 15;
  int bh   = blockIdx.x;
  int h    = bh & (HH - 1);
  int b    = bh >> 3;

  unsigned kldsBase = (unsigned)(size_t)(void*)kT;   // LDS byte offset (aperture low bits)
  const float* kgHead = kg + ((size_t)b * LL * HH + h) * DD;

  // Q fragments (serve as B for S^T): col q = ln, K = d
  v2f qa[16];
  {
    const float* qbase = Qred + ((size_t)bh * PADU + w * 16 + ln) * DD + half * 2;
    #pragma unroll
    for (int kc = 0; kc < 16; ++kc)
      qa[kc] = *(const v2f*)(qbase + kc * 4);
  }

  v8f o[4];
  #pragma unroll
  for (int t = 0; t < 4; ++t) o[t] = (v8f){};
  float mrun = -INFINITY, lsum = 0.f;

  float* pb = pbuf + w * 256;

  // Prefetch K tile 0 via TDM
  if (w == 0) tdm_load_k_tile(kgHead, kldsBase);

  for (int j = 0; j < LL / 16; ++j) {
    __syncthreads();  // all waves done reading kT[(j-1)&1] and vt
    // Prefetch next K tile into the other buffer
    if (w == 0 && j + 1 < LL / 16)
      tdm_load_k_tile(kgHead + (size_t)(j + 1) * 16 * HH * DD, kldsBase + ((j + 1) & 1) * 4096);

    // Stage V tile j transposed (manual: TDM has no transpose)
    {
      int jb = j * 16;
      for (int i = tid; i < 256; i += 96) {
        int key = i >> 4;
        int d4  = (i & 15) << 2;
        float4 val = *(const float4*)(vg + ((size_t)(b * LL + jb + key) * HH + h) * DD + d4);
        vt[(d4 + 0) * VTS + key] = val.x;
        vt[(d4 + 1) * VTS + key] = val.y;
        vt[(d4 + 2) * VTS + key] = val.z;
        vt[(d4 + 3) * VTS + key] = val.w;
      }
    }
    // Tile j's TDM (issued last iteration) must have landed; tile j+1 may stay in flight.
    if (w == 0) {
      if (j + 1 < LL / 16) __builtin_amdgcn_s_wait_tensorcnt(1);
      else                 __builtin_amdgcn_s_wait_tensorcnt(0);
    }
    __syncthreads();

    const float* kTb = kT + (j & 1) * (16 * DD);

    // S^T(16key x 16q) = K(16x64) @ Q^T(64x16); two accumulators halve the RAW chain
    v8f c0 = {}, c1 = {};
    #pragma unroll
    for (int kc = 0; kc < 16; kc += 2) {
      v2f kf0 = *(const v2f*)(&kTb[ln * DD + kc * 4 + half * 2]);
      v2f kf1 = *(const v2f*)(&kTb[ln * DD + (kc + 1) * 4 + half * 2]);
      c0 = __builtin_amdgcn_wmma_f32_16x16x4_f32(false, kf0, false, qa[kc],     (short)0, c0, false, false);
      c1 = __builtin_amdgcn_wmma_f32_16x16x4_f32(false, kf1, false, qa[kc + 1], (short)0, c1, false, false);
    }
    v8f c = c0 + c1;

    // Per-lane online softmax: lane owns query q = w*16+ln
    float tm = c[0];
    #pragma unroll
    for (int vv = 1; vv < 8; ++vv) tm = fmaxf(tm, c[vv]);
    tm = fmaxf(tm, xhalf16(tm));             // combine halves -> tile max
    float mnew = fmaxf(mrun, tm);
    float rs = 0.f;
    #pragma unroll
    for (int vv = 0; vv < 8; ++vv) {
      float p = __expf(c[vv] - mnew);
      c[vv] = p;
      rs += p;
    }
    rs += xhalf16(rs);                       // tile row-sum
    float cr = __expf(mrun - mnew);
    mrun = mnew;
    lsum = lsum * cr + rs;
    #pragma unroll
    for (int t = 0; t < 4; ++t)
      #pragma unroll
      for (int vv = 0; vv < 8; ++vv) o[t][vv] *= cr;

    // P^T -> LDS as [q][key]: contiguous 8 floats per lane (merges to b128 stores)
    {
      float* dst = pb + ln * 16 + 8 * half;
      *(float4*)(dst + 0) = make_float4(c[0], c[1], c[2], c[3]);
      *(float4*)(dst + 4) = make_float4(c[4], c[5], c[6], c[7]);
    }
    __syncthreads();

    // B fragments of P^T: col q = ln, K = key (contiguous pair -> b64)
    v2f pf[4];
    #pragma unroll
    for (int kc2 = 0; kc2 < 4; ++kc2)
      pf[kc2] = *(const v2f*)(&pb[ln * 16 + kc2 * 4 + half * 2]);

    // O^T(64 x 16q) += V^T(64 x 16key) @ P^T(16key x 16q)
    // kc2 outer / t inner: adjacent WMMAs hit different accumulators (no RAW chain)
    #pragma unroll
    for (int kc2 = 0; kc2 < 4; ++kc2) {
      #pragma unroll
      for (int t = 0; t < 4; ++t) {
        v2f vf = *(const v2f*)(&vt[(t * 16 + ln) * VTS + kc2 * 4 + half * 2]);
        o[t] = __builtin_amdgcn_wmma_f32_16x16x4_f32(
                   false, vf, false, pf[kc2], (short)0, o[t], false, false);
      }
    }
  }

  // Epilogue: lane owns query q; O^T element (d = t*16 + v + 8*half, q = ln)
  int qrow = w * 16 + ln;
  if (qrow < NTOP) {
    float inv = 1.f / lsum;
    float* ob = out + (((size_t)b * NTOP + qrow) * HH + h) * DD + 8 * half;
    #pragma unroll
    for (int t = 0; t < 4; ++t) {
      *(float4*)(ob + t * 16)     = make_float4(o[t][0] * inv, o[t][1] * inv,
                                                o[t][2] * inv, o[t][3] * inv);
      *(float4*)(ob + t * 16 + 4) = make_float4(o[t][4] * inv, o[t][5] * inv,
                                                o[t][6] * inv, o[t][7] * inv);
    }
  }
}

extern "C" void kernel_launch(void* const* d_in, const int* in_sizes, int n_in,
                              void* d_out, int out_size, void* d_ws, size_t ws_size,
                              hipStream_t stream) {
  (void)in_sizes; (void)n_in; (void)out_size; (void)ws_size;
  const float* q   = (const float*)d_in[0];
  const float* k   = (const float*)d_in[1];
  const float* v   = (const float*)d_in[2];
  const int*   idx = (const int*)d_in[3];
  float* out = (float*)d_out;

  float* Mv   = (float*)d_ws;                                   // B*H*L floats
  int*   Mtop = (int*)((char*)d_ws + (size_t)BB * HH * LL * 4); // B*H*40 ints
  float* Qred = (float*)((char*)d_ws + (size_t)BB * HH * LL * 4
                                     + (size_t)BB * HH * NTOP * 4); // B*H*48*64

  ps_measure_kernel<<<(BB * HH * LL) / 8, 256, 0, stream>>>(q, k, idx, Mv);
  ps_topk_gather_kernel<<<BB * HH, 256, 0, stream>>>(Mv, q, Mtop, Qred);
  ps_attn_wmma_kernel<<<BB * HH, 96, 0, stream>>>(k, v, Qred, out);
}